// FK_Velocity_Loss_1898375545332
// MI455X (gfx1250) — compile-verified
//
#include <hip/hip_runtime.h>

// ---------------------------------------------------------------------------
// FK translation loss, MI455X (gfx1250).
// Memory-bound streaming reduction: 226 MB in / 8 B out -> ~10us floor @23.3TB/s.
// CDNA5 paths used:
//   * GLOBAL_LOAD_ASYNC_TO_LDS_B128 (ASYNCcnt) double-buffered tile staging
//   * v_wmma_f32_16x16x4_f32 with all-ones B as a wave32 cross-lane reducer
// ---------------------------------------------------------------------------

#define TILE_SAMPLES       256
#define SAMPLE_BYTES       288      // 72 f32
#define SAMPLE_STRIDE_LDS  304      // 288 + 16 pad: stride 76 words -> minimal bank conflicts
#define TILE_BYTES         (TILE_SAMPLES * SAMPLE_STRIDE_LDS)   // 77824 B
#define CHUNKS_PER_SAMPLE  18       // 18 x 16B = 288B
#define CHUNK_ITERS        ((TILE_SAMPLES * CHUNKS_PER_SAMPLE) / 256)  // 18 per thread

typedef __attribute__((ext_vector_type(2))) float v2f;
typedef __attribute__((ext_vector_type(8))) float v8f;
typedef __attribute__((ext_vector_type(4))) int   v4i;

#if defined(__gfx1250__) && \
    __has_builtin(__builtin_amdgcn_global_load_async_to_lds_b128) && \
    __has_builtin(__builtin_amdgcn_s_wait_asynccnt)
#define USE_ASYNC_LDS 1
typedef __attribute__((address_space(1))) v4i GV4i;   // global int4*
typedef __attribute__((address_space(3))) v4i LV4i;   // LDS int4*
#else
#define USE_ASYNC_LDS 0
#endif

// Copy one 256-sample tile (73728B) of pose data into LDS with 16B/sample pad.
// Coalesced: chunk = i*256 + tid walks the tile linearly in 16B units.
__device__ __forceinline__ void tile_copy(const char* __restrict__ g, char* lds,
                                          int tid, long maxOff) {
#pragma unroll
  for (int i = 0; i < CHUNK_ITERS; ++i) {
    int chunk = i * 256 + tid;
    int s = chunk / CHUNKS_PER_SAMPLE;            // sample within tile
    int c = chunk - s * CHUNKS_PER_SAMPLE;        // 16B chunk within sample
    long gOff = (long)chunk * 16;
    gOff = (gOff > maxOff) ? maxOff : gOff;       // tail clamp (select, keeps EXEC full)
    int lOff = s * SAMPLE_STRIDE_LDS + c * 16;
#if USE_ASYNC_LDS
    __builtin_amdgcn_global_load_async_to_lds_b128(
        (GV4i*)(g + gOff), (LV4i*)(lds + lOff), 0, 0);
#else
    *(float4*)(lds + lOff) = *(const float4*)(g + gOff);
#endif
  }
}

template <int N>
__device__ __forceinline__ void wait_async_tiles() {
#if USE_ASYNC_LDS
  __builtin_amdgcn_s_wait_asynccnt(N);   // async loads retire in-order per wave
#endif
}

// 72 floats of one sample: LDS -> registers (18x b128, 16B-aligned stride 304).
__device__ __forceinline__ void load_sample(const char* lds, int tid, float* m) {
  const float4* src = (const float4*)(lds + tid * SAMPLE_STRIDE_LDS);
#pragma unroll
  for (int i = 0; i < 18; ++i) ((float4*)m)[i] = src[i];
}

// Translation of chain M0*M1*M2*M3 for both groups of 4 matrices.
// 3x3 block m[3i+j] (row i, col j); R = [cross(c0,c1) | c0 | c1], t = c2.
// t <- R_k * t + t_k, starting from t = c2 of the last matrix.
__device__ __forceinline__ void fk_translation(const float* m, float out[6]) {
#pragma unroll
  for (int g = 0; g < 2; ++g) {
    const float* q3 = m + (4 * g + 3) * 9;
    float tx = q3[2], ty = q3[5], tz = q3[8];
#pragma unroll
    for (int kk = 2; kk >= 0; --kk) {
      const float* q = m + (4 * g + kk) * 9;
      float xx = q[3] * q[7] - q[6] * q[4];
      float xy = q[6] * q[1] - q[0] * q[7];
      float xz = q[0] * q[4] - q[3] * q[1];
      float nx = xx * tx + q[0] * ty + q[1] * tz + q[2];
      float ny = xy * tx + q[3] * ty + q[4] * tz + q[5];
      float nz = xz * tx + q[6] * ty + q[7] * tz + q[8];
      tx = nx; ty = ny; tz = nz;
    }
    out[3 * g + 0] = tx; out[3 * g + 1] = ty; out[3 * g + 2] = tz;
  }
}

// Wave32 sum via one v_wmma_f32_16x16x4_f32.
// A row m: K0=v(lane m), K1=0, K2=v(lane m+16), K3=0  (documented A layout).
// B = all ones -> layout-independent -> D[m][*] = v(m)+v(m+16).
// C/D layout: VGPR r = rows {r, r+8} split at lane 16; sum 8 regs + shfl_xor(16).
__device__ __forceinline__ float wave_sum_wmma(float v) {
  v2f a; a[0] = v;    a[1] = 0.0f;
  v2f b; b[0] = 1.0f; b[1] = 1.0f;
  v8f c = {0.f, 0.f, 0.f, 0.f, 0.f, 0.f, 0.f, 0.f};
  c = __builtin_amdgcn_wmma_f32_16x16x4_f32(false, a, false, b, (short)0, c,
                                            false, false);
  float s = c[0] + c[1] + c[2] + c[3] + c[4] + c[5] + c[6] + c[7];
  s += __shfl_xor(s, 16, 32);
  return s;   // full 32-lane sum, present in every lane
}

__global__ __launch_bounds__(256) void fk_loss_stage1(
    const float* __restrict__ outP, const float* __restrict__ gtP,
    const float* __restrict__ prevP, float* __restrict__ partials, long B) {
  extern __shared__ char smem[];
  char*  buf0  = smem;                         // 77824 B
  char*  buf1  = smem + TILE_BYTES;            // 77824 B (155648 total, < 160KB)
  float* wpart = (float*)(smem + 2 * TILE_BYTES);

  const int  tid     = threadIdx.x;
  const long sBase   = (long)blockIdx.x * TILE_SAMPLES;
  const long tileOff = sBase * (long)SAMPLE_BYTES;
  const long maxOff  = B * (long)SAMPLE_BYTES - tileOff - 16;

  // Fill the async pipe: 36 loads in flight per wave (ASYNCcnt max 63).
  tile_copy((const char*)outP + tileOff, buf0, tid, maxOff);
  tile_copy((const char*)gtP  + tileOff, buf1, tid, maxOff);

  wait_async_tiles<CHUNK_ITERS>();   // oldest 18 done -> out_pose tile resident
  __syncthreads();

  float m[72], fkO[6], fkG[6], fkP[6];
  load_sample(buf0, tid, m);
  fk_translation(m, fkO);
  __syncthreads();                   // everyone done reading buf0

  tile_copy((const char*)prevP + tileOff, buf0, tid, maxOff);  // overlaps fkG

  wait_async_tiles<CHUNK_ITERS>();   // gt_pose tile resident
  __syncthreads();
  load_sample(buf1, tid, m);
  fk_translation(m, fkG);

  wait_async_tiles<0>();             // gt_prev tile resident
  __syncthreads();
  load_sample(buf0, tid, m);
  fk_translation(m, fkP);

  // Per-sample squared errors; masked (not branched) so EXEC stays all-ones
  // for the WMMA reduction below.
  float mask = ((sBase + tid) < B) ? 1.0f : 0.0f;
  float pos = 0.0f, vel = 0.0f;
#pragma unroll
  for (int i = 0; i < 6; ++i) {
    float dp = fkO[i] - fkG[i];
    pos += dp * dp;
    float dv = (fkO[i] - fkP[i]) - (fkG[i] - fkP[i]);
    vel += dv * dv;
  }
  pos *= mask; vel *= mask;

  float wp = wave_sum_wmma(pos);
  float wv = wave_sum_wmma(vel);

  int wave = tid >> 5, lane = tid & 31;
  if (lane == 0) { wpart[2 * wave] = wp; wpart[2 * wave + 1] = wv; }
  __syncthreads();
  if (tid == 0) {
    float P = 0.0f, V = 0.0f;
#pragma unroll
    for (int w = 0; w < 8; ++w) { P += wpart[2 * w]; V += wpart[2 * w + 1]; }
    partials[2 * blockIdx.x]     = P;
    partials[2 * blockIdx.x + 1] = V;
  }
}

__global__ __launch_bounds__(256) void fk_loss_stage2(
    const float* __restrict__ partials, float* __restrict__ out,
    int nblocks, float inv) {
  __shared__ float wpart[16];
  const int tid = threadIdx.x;
  float p = 0.0f, v = 0.0f;
  for (int i = tid; i < nblocks; i += 256) {
    p += partials[2 * i];
    v += partials[2 * i + 1];
  }
  float wp = wave_sum_wmma(p);
  float wv = wave_sum_wmma(v);
  int wave = tid >> 5, lane = tid & 31;
  if (lane == 0) { wpart[2 * wave] = wp; wpart[2 * wave + 1] = wv; }
  __syncthreads();
  if (tid == 0) {
    float P = 0.0f, V = 0.0f;
#pragma unroll
    for (int w = 0; w < 8; ++w) { P += wpart[2 * w]; V += wpart[2 * w + 1]; }
    out[0] = P * inv;   // pos_loss = mean over B*2*3 values
    out[1] = V * inv;   // vel_loss
  }
}

extern "C" void kernel_launch(void* const* d_in, const int* in_sizes, int n_in,
                              void* d_out, int out_size, void* d_ws, size_t ws_size,
                              hipStream_t stream) {
  const float* outP  = (const float*)d_in[0];
  const float* gtP   = (const float*)d_in[1];
  const float* prevP = (const float*)d_in[2];
  long B = (long)in_sizes[0] / 72;                       // 262144
  int nblocks = (int)((B + TILE_SAMPLES - 1) / TILE_SAMPLES);  // 1024
  float* partials = (float*)d_ws;                        // nblocks*2 floats

  size_t shmem = 2 * (size_t)TILE_BYTES + 64;            // 155712 B dynamic LDS
  fk_loss_stage1<<<nblocks, 256, shmem, stream>>>(outP, gtP, prevP, partials, B);

  float inv = 1.0f / (6.0f * (float)B);
  fk_loss_stage2<<<1, 256, 0, stream>>>(partials, (float*)d_out, nblocks, inv);
}